// SSIM_14439680049421
// MI455X (gfx1250) — compile-verified
//
#include <hip/hip_runtime.h>

typedef __attribute__((ext_vector_type(2))) float v2f;
typedef __attribute__((ext_vector_type(8))) float v8f;

#define IMG_H 512
#define IMG_W 512
#define NIMG  32
#define TR    64      // output rows per block
#define TC    32      // output cols per block
#define HALO  5
#define XROWS 80      // 74 halo rows padded to 5 strips of 16
#define XPIT  44      // 42 halo cols padded (bank-conflict-free pitch)
#define TROWS 80
#define TPIT  44
#define C1F   1.0e-4f
#define C2F   9.0e-4f

// 7 chained K=4 chunks cover the padded K=28 (>= 26 needed for 11-tap band)
#define WMMA4(A, B, C) \
    __builtin_amdgcn_wmma_f32_16x16x4_f32(false, (A), false, (B), (short)0, (C), false, false)

__global__ __launch_bounds__(256)
void ssim_main(const float* __restrict__ logits, const float* __restrict__ gts,
               float* __restrict__ partial) {
    __shared__ float sX[XROWS][XPIT];
    __shared__ float sY[XROWS][XPIT];
    __shared__ float sT[5][TROWS][TPIT];
    __shared__ float sG[11];
    __shared__ float sRed[256];

    const int tid  = threadIdx.x;
    const int lane = tid & 31;
    const int wave = tid >> 5;
    const int n15  = lane & 15;
    const int hi2  = (lane >= 16) ? 2 : 0;

    // Gaussian 1-D weights, double precision like numpy, once per block.
    if (tid == 0) {
        double t[11], s = 0.0;
        for (int i = 0; i < 11; ++i) { double d = (double)(i - 5); t[i] = exp(-(d * d) / 4.5); s += t[i]; }
        for (int i = 0; i < 11; ++i) sG[i] = (float)(t[i] / s);
    }
    __syncthreads();

    // Per-lane constant banded fragment. Same formula serves stage-1 B and
    // stage-2 A: value = w[(4*kk + v + hi2) - n15] inside the band, else 0.
    float cF[7][2];
#pragma unroll
    for (int kk = 0; kk < 7; ++kk)
#pragma unroll
        for (int v = 0; v < 2; ++v) {
            int q = 4 * kk + v + hi2 - n15;
            cF[kk][v] = (q >= 0 && q <= 10) ? sG[q] : 0.0f;
        }

    // ---- Stage 0: cooperative load of the full padded region (finite everywhere)
    const int    n    = blockIdx.z;
    const int    r0   = blockIdx.y * TR;
    const int    c0   = blockIdx.x * TC;
    const size_t base = (size_t)n * (size_t)(IMG_H * IMG_W);
    for (int idx = tid; idx < XROWS * XPIT; idx += 256) {
        int i = idx / XPIT, j = idx - i * XPIT;
        int R = r0 - HALO + i, C = c0 - HALO + j;
        float xv = 0.0f, yv = 0.0f;
        if (R >= 0 && R < IMG_H && C >= 0 && C < IMG_W) {
            size_t o  = base + (size_t)R * IMG_W + (size_t)C;
            float  lg = logits[o];
            xv = 1.0f / (1.0f + __expf(-lg));   // sigmoid
            yv = gts[o];
        }
        sX[i][j] = xv;
        sY[i][j] = yv;
    }
    __syncthreads();

    // ---- Stage 1: horizontal 11-tap conv as WMMA, data in A, band in B.
    // Jobs: strip s in [0,5) x coltile ct in [0,2) = 10 jobs over 8 waves.
    for (int job = wave; job < 10; job += 8) {       // wave-uniform branch: EXEC stays full
        const int s  = job >> 1;
        const int ct = job & 1;
        const int rowA = 16 * s + n15;               // A: lane = row M
        v2f xa[7], ya[7];
#pragma unroll
        for (int kk = 0; kk < 7; ++kk)
#pragma unroll
            for (int v = 0; v < 2; ++v) {
                int col = 16 * ct + 4 * kk + v + hi2;   // max 43 < XPIT
                xa[kk][v] = sX[rowA][col];
                ya[kk][v] = sY[rowA][col];
            }
        v8f a0 = {}, a1 = {}, a2 = {}, a3 = {}, a4 = {};
#pragma unroll
        for (int kk = 0; kk < 7; ++kk) {
            v2f b = {};
            b[0] = cF[kk][0]; b[1] = cF[kk][1];
            v2f x = xa[kk], y = ya[kk];
            v2f xx = x * x, yy = y * y, xy = x * y;
            a0 = WMMA4(x,  b, a0);
            a1 = WMMA4(y,  b, a1);
            a2 = WMMA4(xx, b, a2);
            a3 = WMMA4(yy, b, a3);
            a4 = WMMA4(xy, b, a4);
        }
        const int colD = 16 * ct + n15;              // D: lane = column N
#pragma unroll
        for (int v = 0; v < 8; ++v) {
            int rowD = 16 * s + ((lane < 16) ? v : 8 + v);
            sT[0][rowD][colD] = a0[v];
            sT[1][rowD][colD] = a1[v];
            sT[2][rowD][colD] = a2[v];
            sT[3][rowD][colD] = a3[v];
            sT[4][rowD][colD] = a4[v];
        }
    }
    __syncthreads();

    // ---- Stage 2: vertical conv as WMMA, band in A, data in B; then SSIM map.
    float localSum = 0.0f;
    {
        const int m  = wave >> 1;                    // 8 jobs, exactly 1 per wave
        const int ct = wave & 1;
        v8f a0 = {}, a1 = {}, a2 = {}, a3 = {}, a4 = {};
#pragma unroll
        for (int kk = 0; kk < 7; ++kk) {
            v2f a = {};
            a[0] = cF[kk][0]; a[1] = cF[kk][1];
            v2f b0 = {}, b1 = {}, b2 = {}, b3 = {}, b4 = {};
#pragma unroll
            for (int v = 0; v < 2; ++v) {
                int row = 16 * m + 4 * kk + v + hi2;    // max 75 < TROWS
                int col = 16 * ct + n15;                // B: lane = column N
                b0[v] = sT[0][row][col];
                b1[v] = sT[1][row][col];
                b2[v] = sT[2][row][col];
                b3[v] = sT[3][row][col];
                b4[v] = sT[4][row][col];
            }
            a0 = WMMA4(a, b0, a0);
            a1 = WMMA4(a, b1, a1);
            a2 = WMMA4(a, b2, a2);
            a3 = WMMA4(a, b3, a3);
            a4 = WMMA4(a, b4, a4);
        }
#pragma unroll
        for (int v = 0; v < 8; ++v) {
            float mu1 = a0[v], mu2 = a1[v], ex2 = a2[v], ey2 = a3[v], exy = a4[v];
            float mu1s = mu1 * mu1, mu2s = mu2 * mu2, mu12 = mu1 * mu2;
            float s1 = ex2 - mu1s, s2 = ey2 - mu2s, s12 = exy - mu12;
            float num = (2.0f * mu12 + C1F) * (2.0f * s12 + C2F);
            float den = (mu1s + mu2s + C1F) * (s1 + s2 + C2F);
            localSum += num / den;
        }
    }

    // ---- Deterministic block reduction
    sRed[tid] = localSum;
    __syncthreads();
    for (int off = 128; off > 0; off >>= 1) {
        if (tid < off) sRed[tid] += sRed[tid + off];
        __syncthreads();
    }
    if (tid == 0) {
        int bidx = (blockIdx.z * gridDim.y + blockIdx.y) * gridDim.x + blockIdx.x;
        partial[bidx] = sRed[0];
    }
}

__global__ __launch_bounds__(256)
void ssim_final(const float* __restrict__ partial, int np, float* __restrict__ out) {
    __shared__ float sRed[256];
    float s = 0.0f;
    for (int i = threadIdx.x; i < np; i += 256) s += partial[i];
    sRed[threadIdx.x] = s;
    __syncthreads();
    for (int off = 128; off > 0; off >>= 1) {
        if (threadIdx.x < off) sRed[threadIdx.x] += sRed[threadIdx.x + off];
        __syncthreads();
    }
    if (threadIdx.x == 0)
        out[0] = 1.0f - sRed[0] / (float)((size_t)NIMG * IMG_H * IMG_W);
}

extern "C" void kernel_launch(void* const* d_in, const int* in_sizes, int n_in,
                              void* d_out, int out_size, void* d_ws, size_t ws_size,
                              hipStream_t stream) {
    (void)in_sizes; (void)n_in; (void)out_size; (void)ws_size;
    const float* logits = (const float*)d_in[0];
    const float* gts    = (const float*)d_in[1];
    // d_in[2] (window) is outer(g,g) of the fixed Gaussian; weights are
    // regenerated in-kernel in double precision, matching numpy's construction.
    float* partial = (float*)d_ws;

    dim3 grid(IMG_W / TC, IMG_H / TR, NIMG);   // 16 x 8 x 32 = 4096 blocks
    ssim_main<<<grid, 256, 0, stream>>>(logits, gts, partial);

    const int np = (IMG_W / TC) * (IMG_H / TR) * NIMG;
    ssim_final<<<1, 256, 0, stream>>>(partial, np, (float*)d_out);
}